// GCNLayer_62362925138833
// MI455X (gfx1250) — compile-verified
//
#include <hip/hip_runtime.h>

#define N_NODES 40000
#define N_EDGES 640000
#define DD      128

typedef float v2f __attribute__((ext_vector_type(2)));
typedef float v8f __attribute__((ext_vector_type(8)));

// ---------------------------------------------------------------- zero ws
__global__ void zero_ws(float* __restrict__ p, int total) {
  int i = blockIdx.x * blockDim.x + threadIdx.x;
  int stride = gridDim.x * blockDim.x;
  for (; i < total; i += stride) p[i] = 0.0f;
}

// ---------------------------------------------------------------- degrees
__global__ void degrees(const int* __restrict__ src, const int* __restrict__ dst,
                        float* __restrict__ deg_out, float* __restrict__ deg_in) {
  int e = blockIdx.x * blockDim.x + threadIdx.x;
  if (e < N_EDGES) {
    atomicAdd(&deg_out[src[e]], 1.0f);
    atomicAdd(&deg_in[dst[e]], 1.0f);
  }
}

// ---------------------------------------------------------------- edge scatter
// One wave32 per edge: lane l moves h[src][4l..4l+3] (float4 => 32x16B = 512B row),
// scaled by mask[e] * rsqrt(max(out_deg[src],1)), atomically added into agg[dst].
__global__ void scatter(const float* __restrict__ h, const float* __restrict__ mask,
                        const int* __restrict__ src, const int* __restrict__ dst,
                        const float* __restrict__ deg_out, float* __restrict__ agg) {
  int wid  = threadIdx.x >> 5;    // wave32
  int lane = threadIdx.x & 31;
  int e = blockIdx.x * 8 + wid;   // 8 waves per 256-thread block
  if (e >= N_EDGES) return;
  int s = src[e];
  int d = dst[e];
  float scale = mask[e] * rsqrtf(fmaxf(deg_out[s], 1.0f));
  const float4* h4 = (const float4*)(h + (size_t)s * DD);
  float4 v = h4[lane];
  float* ap = agg + (size_t)d * DD + lane * 4;
  atomicAdd(ap + 0, v.x * scale);
  atomicAdd(ap + 1, v.y * scale);
  atomicAdd(ap + 2, v.z * scale);
  atomicAdd(ap + 3, v.w * scale);
}

// ---------------------------------------------------------------- GEMM + epilogue
// out = h*out_deg^-0.5 + relu((agg @ W + b) * in_deg^-0.5)
// 4 waves / block; each wave computes a 16x128 output slab with
// v_wmma_f32_16x16x4_f32 (8 column tiles x 32 K-steps).
__global__ void __launch_bounds__(128)
gemm_epilogue(const float* __restrict__ agg, const float* __restrict__ W,
              const float* __restrict__ bias, const float* __restrict__ h,
              const float* __restrict__ deg_out, const float* __restrict__ deg_in,
              float* __restrict__ out) {
  __shared__ float Atile[4][16 * DD];   // 32 KB: private 8KB slab per wave

  const int wid  = threadIdx.x >> 5;
  const int lane = threadIdx.x & 31;
  const int tile = blockIdx.x * 4 + wid;   // 2500 tiles, 625 blocks: exact
  const int row0 = tile * 16;

  float* As = Atile[wid];

  // Stage 16x128 A tile into LDS, fully coalesced (one 512B row per iteration).
  for (int r = 0; r < 16; ++r) {
    const float4* srcp = (const float4*)(agg + (size_t)(row0 + r) * DD);
    *(float4*)(&As[r * DD + lane * 4]) = srcp[lane];
  }

  const int half = lane >> 4;   // lane half selects K=+2 per ISA 16x4 layout
  const int n    = lane & 15;   // M index for A frags / N index for B,C,D frags

  v8f zero = {};
  v8f acc[8];
#pragma unroll
  for (int ct = 0; ct < 8; ++ct) acc[ct] = zero;

  for (int k = 0; k < DD; k += 4) {
    const int ks = k + 2 * half;
    // A fragment (16x4 f32): v0 = A[m][ks], v1 = A[m][ks+1], m = lane&15
    v2f a;
    a.x = As[n * DD + ks];
    a.y = As[n * DD + ks + 1];
#pragma unroll
    for (int ct = 0; ct < 8; ++ct) {
      // B fragment (4x16 f32): v0 = W[ks][col], v1 = W[ks+1][col]
      v2f bf;
      bf.x = W[(size_t)ks * DD + ct * 16 + n];
      bf.y = W[(size_t)(ks + 1) * DD + ct * 16 + n];
      acc[ct] = __builtin_amdgcn_wmma_f32_16x16x4_f32(
          false, a, false, bf, (short)0, acc[ct], false, false);
    }
  }

  // Epilogue. C/D layout: VGPR j -> row = row0 + j + 8*half, col = ct*16 + n.
#pragma unroll
  for (int j = 0; j < 8; ++j) {
    const int row = row0 + j + 8 * half;
    const float isd_in  = rsqrtf(fmaxf(deg_in[row],  1.0f));
    const float isd_out = rsqrtf(fmaxf(deg_out[row], 1.0f));
#pragma unroll
    for (int ct = 0; ct < 8; ++ct) {
      const int col = ct * 16 + n;
      float val = (acc[ct][j] + bias[col]) * isd_in;
      val = fmaxf(val, 0.0f) + h[(size_t)row * DD + col] * isd_out;
      out[(size_t)row * DD + col] = val;
    }
  }
}

// ---------------------------------------------------------------- launcher
extern "C" void kernel_launch(void* const* d_in, const int* in_sizes, int n_in,
                              void* d_out, int out_size, void* d_ws, size_t ws_size,
                              hipStream_t stream) {
  const float* h    = (const float*)d_in[0];
  const float* mask = (const float*)d_in[1];
  const float* W    = (const float*)d_in[2];
  const float* b    = (const float*)d_in[3];
  const int*   src  = (const int*)d_in[4];
  const int*   dst  = (const int*)d_in[5];
  float* out = (float*)d_out;

  float* ws      = (float*)d_ws;
  float* deg_out = ws;                 // [N]
  float* deg_in  = ws + N_NODES;       // [N]
  float* agg     = ws + 2 * N_NODES;   // [N*D], 16B-aligned (2N floats = 320000B)

  const int total_ws = 2 * N_NODES + N_NODES * DD;

  zero_ws<<<2048, 256, 0, stream>>>(ws, total_ws);
  degrees<<<(N_EDGES + 255) / 256, 256, 0, stream>>>(src, dst, deg_out, deg_in);
  scatter<<<N_EDGES / 8, 256, 0, stream>>>(h, mask, src, dst, deg_out, agg);
  gemm_epilogue<<<(N_NODES / 16) / 4, 128, 0, stream>>>(agg, W, b, h,
                                                        deg_out, deg_in, out);
}